// Loss_reference_neg_85942295593595
// MI455X (gfx1250) — compile-verified
//
#include <hip/hip_runtime.h>

// ---------------- problem constants (match reference) ----------------
#define B_       256
#define L_       256
#define D_       300
#define C_       5
#define NREF_    50
#define ITERS_   50
#define EPS_     0.0025f        // blur^2 = 0.05^2
#define INV_EPS_ 400.0f
#define NEG_BIG_ (-1.0e9f)
#define BLOG_    (-3.912023005428146f)   // log(1/NREF)
#define MARGIN_  10.0f
#define CSTRIDE_ 260            // padded LDS row stride for Cxx (bank-safe)

typedef float v2f __attribute__((ext_vector_type(2)));
typedef float v8f __attribute__((ext_vector_type(8)));

// fp32 WMMA: D(16x16,f32) += A(16x4,f32) * B(4x16,f32)
__device__ __forceinline__ v8f wmma_f32x4(v2f a, v2f b, v8f c) {
  return __builtin_amdgcn_wmma_f32_16x16x4_f32(false, a, false, b, (short)0, c,
                                               false, false);
}

__device__ __forceinline__ float wave_max(float v) {
#pragma unroll
  for (int o = 16; o > 0; o >>= 1) v = fmaxf(v, __shfl_xor(v, o, 32));
  return v;
}
__device__ __forceinline__ float wave_sum(float v) {
#pragma unroll
  for (int o = 16; o > 0; o >>= 1) v += __shfl_xor(v, o, 32);
  return v;
}

// ---------------------------------------------------------------------
// K1: per-class c: yn2[c][j], Cyy in LDS, 50 q-iterations -> q[c][j]
// ---------------------------------------------------------------------
__global__ __launch_bounds__(256) void k_q(const float* __restrict__ t0,
                                           float* __restrict__ q_ws,
                                           float* __restrict__ yn2_ws) {
  const int c = blockIdx.x;
  const int tid = threadIdx.x;
  __shared__ float s_cyy[NREF_ * NREF_];
  __shared__ float s_yn2[64], s_h[64], s_q[64];
  const float* y = t0 + (size_t)c * NREF_ * D_;

  if (tid < NREF_) {
    float acc = 0.f;
    const float* yr = y + tid * D_;
    for (int d = 0; d < D_; ++d) acc = fmaf(yr[d], yr[d], acc);
    s_yn2[tid] = acc;
    yn2_ws[c * NREF_ + tid] = acc;
  }
  __syncthreads();

  for (int idx = tid; idx < NREF_ * NREF_; idx += 256) {
    int i = idx / NREF_, j = idx % NREF_;
    const float* yi = y + i * D_;
    const float* yj = y + j * D_;
    float dot = 0.f;
    for (int d = 0; d < D_; ++d) dot = fmaf(yi[d], yj[d], dot);
    s_cyy[idx] = 0.5f * (s_yn2[i] + s_yn2[j]) - dot;
  }
  __syncthreads();

  for (int it = 0; it <= ITERS_; ++it) {
    if (tid < NREF_)
      s_h[tid] = (it == 0) ? BLOG_ : (BLOG_ + s_q[tid] * INV_EPS_);
    __syncthreads();
    if (tid < NREF_) {
      const float* cr = &s_cyy[tid * NREF_];
      float m = -3.0e38f;
      for (int j = 0; j < NREF_; ++j) m = fmaxf(m, s_h[j] - cr[j] * INV_EPS_);
      float s = 0.f;
      for (int j = 0; j < NREF_; ++j) s += __expf(s_h[j] - cr[j] * INV_EPS_ - m);
      float sm = -EPS_ * (m + __logf(s));
      s_q[tid] = (it == 0) ? sm : 0.5f * (s_q[tid] + sm);
    }
    __syncthreads();
  }
  if (tid < NREF_) q_ws[c * NREF_ + tid] = s_q[tid];
}

// ---------------------------------------------------------------------
// K2 (fused): per-b: xn2, alog, Cxx (256x256) via fp32 WMMA kept entirely
// in LDS (~270KB of the 320KB WGP LDS), then 1+50 p-iterations on it.
// ---------------------------------------------------------------------
__global__ __launch_bounds__(256) void k_p_fused(
    const float* __restrict__ anchor, const float* __restrict__ weight,
    float* __restrict__ xn2_ws, float* __restrict__ alog_ws,
    float* __restrict__ p_ws) {
  const int b = blockIdx.x;
  const int tid = threadIdx.x, lane = tid & 31, w = tid >> 5;
  __shared__ float s_C[L_ * CSTRIDE_];  // 256 x 260 fp32 = 266240 B
  __shared__ float s_xn2[L_], s_alog[L_], s_h[L_], s_p[L_];
  const float* x = anchor + (size_t)b * L_ * D_;

  {  // row norms + log weights
    float acc = 0.f;
    const float* xr = x + tid * D_;
    for (int d = 0; d < D_; ++d) acc = fmaf(xr[d], xr[d], acc);
    s_xn2[tid] = acc;
    xn2_ws[(size_t)b * L_ + tid] = acc;
    float wt = weight[(size_t)b * L_ + tid];
    float al = (wt > 0.f) ? __logf(wt) : NEG_BIG_;
    s_alog[tid] = al;
    alog_ws[(size_t)b * L_ + tid] = al;
  }
  __syncthreads();

  // Gram: 16x16 output tiles, 16 mt x 16 nt = 256 tiles, 32 per wave
  const int ksub = (lane >> 4) * 2;
  for (int t = w * 32; t < w * 32 + 32; ++t) {
    const int mt = t >> 4, nt = t & 15;
    const float* pa = x + (mt * 16 + (lane & 15)) * D_ + ksub;
    const float* pb = x + (nt * 16 + (lane & 15)) * D_ + ksub;
    v8f acc = {0.f, 0.f, 0.f, 0.f, 0.f, 0.f, 0.f, 0.f};
    for (int k = 0; k < D_; k += 4) {
      v2f a = {pa[k], pa[k + 1]};
      v2f bb = {pb[k], pb[k + 1]};
      acc = wmma_f32x4(a, bb, acc);
    }
    const int nj = nt * 16 + (lane & 15);
    const int mbase = mt * 16 + (lane >> 4) * 8;
    const float ynj = s_xn2[nj];
#pragma unroll
    for (int r = 0; r < 8; ++r) {
      const int mi = mbase + r;
      s_C[mi * CSTRIDE_ + nj] = 0.5f * (s_xn2[mi] + ynj) - acc[r];
    }
  }
  __syncthreads();

  // 1 init + 50 symmetric iterations, all reads from LDS
  for (int it = 0; it <= ITERS_; ++it) {
    s_h[tid] = (it == 0) ? s_alog[tid] : (s_alog[tid] + s_p[tid] * INV_EPS_);
    __syncthreads();
    // wave-per-row, lanes split the 256 columns (stride-1 banks)
    for (int rr = 0; rr < 32; ++rr) {
      const int row = w * 32 + rr;
      const float* cr = &s_C[row * CSTRIDE_];
      float v[8];
      float m = -3.0e38f;
#pragma unroll
      for (int k2 = 0; k2 < 8; ++k2) {
        const int j = lane + 32 * k2;
        v[k2] = s_h[j] - cr[j] * INV_EPS_;
        m = fmaxf(m, v[k2]);
      }
      m = wave_max(m);
      float s = 0.f;
#pragma unroll
      for (int k2 = 0; k2 < 8; ++k2) s += __expf(v[k2] - m);
      s = wave_sum(s);
      const float sm = -EPS_ * (m + __logf(s));
      if (lane == 0) s_p[row] = (it == 0) ? sm : 0.5f * (s_p[row] + sm);
    }
    __syncthreads();
  }
  p_ws[(size_t)b * L_ + tid] = s_p[tid];
}

// ---------------------------------------------------------------------
// K3: per-(b,c): Cxy via WMMA into LDS, 50 f/g iterations, dist
// ---------------------------------------------------------------------
__global__ __launch_bounds__(256) void k_fg(const float* __restrict__ anchor,
                                            const float* __restrict__ weight,
                                            const float* __restrict__ t0,
                                            const float* __restrict__ p_ws,
                                            const float* __restrict__ q_ws,
                                            const float* __restrict__ xn2_ws,
                                            const float* __restrict__ yn2_ws,
                                            const float* __restrict__ alog_ws,
                                            float* __restrict__ dists) {
  const int c = blockIdx.x;
  const int b = blockIdx.y;
  const int tid = threadIdx.x, lane = tid & 31, w = tid >> 5;
  __shared__ float s_C[L_ * NREF_];  // row-major 256x50, bank-conflict free
  __shared__ float s_xn2[L_], s_alog[L_], s_f[L_], s_hA[L_];
  __shared__ float s_yn2[64], s_g[64], s_hB[64];
  __shared__ float s_red[8];
  const float* x = anchor + (size_t)b * L_ * D_;
  const float* y = t0 + (size_t)c * NREF_ * D_;

  s_xn2[tid] = xn2_ws[(size_t)b * L_ + tid];
  s_alog[tid] = alog_ws[(size_t)b * L_ + tid];
  if (tid < NREF_) s_yn2[tid] = yn2_ws[c * NREF_ + tid];
  __syncthreads();

  // Cxy: 16 mt x 4 nt = 64 tiles, 8 per wave
  const int ksub = (lane >> 4) * 2;
  for (int t = w * 8; t < w * 8 + 8; ++t) {
    const int mt = t >> 2, nt = t & 3;
    const int colN = nt * 16 + (lane & 15);
    const int rowY = (colN < NREF_) ? colN : 0;  // clamp OOB ref rows
    const int rowA = mt * 16 + (lane & 15);
    const float* pa = x + rowA * D_ + ksub;
    const float* pb = y + rowY * D_ + ksub;
    v8f acc = {0.f, 0.f, 0.f, 0.f, 0.f, 0.f, 0.f, 0.f};
    for (int k = 0; k < D_; k += 4) {
      v2f a = {pa[k], pa[k + 1]};
      v2f bb = {pb[k], pb[k + 1]};
      acc = wmma_f32x4(a, bb, acc);
    }
    if (colN < NREF_) {
      const float yn = s_yn2[colN];
      const int mbase = mt * 16 + (lane >> 4) * 8;
#pragma unroll
      for (int r = 0; r < 8; ++r)
        s_C[(mbase + r) * NREF_ + colN] = 0.5f * (s_xn2[mbase + r] + yn) - acc[r];
    }
  }
  __syncthreads();

  for (int it = 0; it <= ITERS_; ++it) {
    if (it == 0) {
      if (tid < NREF_) s_hB[tid] = BLOG_;
      s_hA[tid] = s_alog[tid];
    } else {
      if (tid < NREF_) s_hB[tid] = BLOG_ + s_g[tid] * INV_EPS_;
      s_hA[tid] = s_alog[tid] + s_f[tid] * INV_EPS_;
    }
    __syncthreads();

    // f-pass: one thread per x-row, serial over 50 cols (stride-50 LDS: no conflicts)
    {
      const float* cr = &s_C[tid * NREF_];
      float m = -3.0e38f;
      for (int j = 0; j < NREF_; ++j) m = fmaxf(m, s_hB[j] - cr[j] * INV_EPS_);
      float s = 0.f;
      for (int j = 0; j < NREF_; ++j)
        s += __expf(s_hB[j] - cr[j] * INV_EPS_ - m);
      const float sm = -EPS_ * (m + __logf(s));
      s_f[tid] = (it == 0) ? sm : 0.5f * (s_f[tid] + sm);
    }
    // g-pass: wave per column, lanes split the 256 rows
    for (int j = w; j < NREF_; j += 8) {
      float v[8];
      float m = -3.0e38f;
#pragma unroll
      for (int k2 = 0; k2 < 8; ++k2) {
        const int i = lane + 32 * k2;
        v[k2] = s_hA[i] - s_C[i * NREF_ + j] * INV_EPS_;
        m = fmaxf(m, v[k2]);
      }
      m = wave_max(m);
      float s = 0.f;
#pragma unroll
      for (int k2 = 0; k2 < 8; ++k2) s += __expf(v[k2] - m);
      s = wave_sum(s);
      const float sm = -EPS_ * (m + __logf(s));
      if (lane == 0) s_g[j] = (it == 0) ? sm : 0.5f * (s_g[j] + sm);
    }
    __syncthreads();
  }

  // dist = sum_i a_i (f_i - p_i) + sum_j (1/NREF)(g_j - q_j)
  float contrib =
      weight[(size_t)b * L_ + tid] * (s_f[tid] - p_ws[(size_t)b * L_ + tid]);
  if (tid < NREF_)
    contrib += (1.0f / NREF_) * (s_g[tid] - q_ws[c * NREF_ + tid]);
  float sw = wave_sum(contrib);
  if (lane == 0) s_red[w] = sw;
  __syncthreads();
  if (tid == 0) {
    float tot = 0.f;
#pragma unroll
    for (int i = 0; i < 8; ++i) tot += s_red[i];
    dists[b * C_ + c] = tot;
  }
}

// ---------------------------------------------------------------------
// K4: margin loss + mean
// ---------------------------------------------------------------------
__global__ __launch_bounds__(256) void k_loss(const float* __restrict__ dists,
                                              const int* __restrict__ grade,
                                              float* __restrict__ out) {
  const int b = threadIdx.x;
  const int g = grade[b];
  const float pos = dists[b * C_ + g];
  float acc = 0.f;
#pragma unroll
  for (int k = 0; k < C_; ++k)
    if (k != g) acc += fmaxf(pos - dists[b * C_ + k] + MARGIN_, 0.f);
  acc *= 0.2f;  // / 5
  __shared__ float red[8];
  float s = wave_sum(acc);
  if ((threadIdx.x & 31) == 0) red[threadIdx.x >> 5] = s;
  __syncthreads();
  if (threadIdx.x == 0) {
    float t = 0.f;
#pragma unroll
    for (int i = 0; i < 8; ++i) t += red[i];
    out[0] = t / (float)B_;
  }
}

// ---------------------------------------------------------------------
extern "C" void kernel_launch(void* const* d_in, const int* in_sizes, int n_in,
                              void* d_out, int out_size, void* d_ws,
                              size_t ws_size, hipStream_t stream) {
  (void)in_sizes; (void)n_in; (void)out_size; (void)ws_size;
  const float* anchor = (const float*)d_in[0];
  const float* weight = (const float*)d_in[1];
  const float* t0     = (const float*)d_in[2];
  // d_in[3] anchor_out_neg: unused by reference
  // d_in[4] length_anchor:  mask already baked into weight
  const int* grade = (const int*)d_in[5];
  float* out = (float*)d_out;

  float* ws    = (float*)d_ws;
  float* q     = ws;                        //   256 (250 used)
  float* yn2   = ws + 256;                  //   256 (250 used)
  float* p     = ws + 512;                  // 65536
  float* alog  = p + (size_t)B_ * L_;       // 65536
  float* xn2   = alog + (size_t)B_ * L_;    // 65536
  float* dists = xn2 + (size_t)B_ * L_;     //  2048 (1280 used)
  // total ws: ~199K floats (~0.8 MB) — Cxx now lives in LDS

  k_q<<<C_, 256, 0, stream>>>(t0, q, yn2);
  k_p_fused<<<B_, 256, 0, stream>>>(anchor, weight, xn2, alog, p);
  dim3 grid(C_, B_);
  k_fg<<<grid, 256, 0, stream>>>(anchor, weight, t0, p, q, xn2, yn2, alog,
                                 dists);
  k_loss<<<1, 256, 0, stream>>>(dists, grade, out);
}